// CreditRiskGNN_80925773791603
// MI455X (gfx1250) — compile-verified
//
#include <hip/hip_runtime.h>
#include <hip/hip_bf16.h>

typedef __attribute__((ext_vector_type(2))) float v2f;
typedef __attribute__((ext_vector_type(8))) float v8f;

#define WAVE 32

// ---------------------------------------------------------------------------
// deg[i] = 1.0 (self loop)
__global__ void k_deg_init(float* deg, int n) {
    int i = blockIdx.x * blockDim.x + threadIdx.x;
    if (i < n) deg[i] = 1.0f;
}

// deg[dst[e]] += 1 for every edge
__global__ void k_deg_accum(const long long* __restrict__ dst, float* deg, int e) {
    int i = blockIdx.x * blockDim.x + threadIdx.x;
    if (i < e) atomicAdd(&deg[(int)dst[i]], 1.0f);
}

// dinv[i] = rsqrt(deg[i])   (deg >= 1 always, self-loop)
__global__ void k_dinv(float* deg, int n) {
    int i = blockIdx.x * blockDim.x + threadIdx.x;
    if (i < n) deg[i] = rsqrtf(deg[i]);
}

// norm[e] = dinv[src]*dinv[dst]
__global__ void k_norm(const long long* __restrict__ src, const long long* __restrict__ dst,
                       const float* __restrict__ dinv, float* __restrict__ norm, int e) {
    int i = blockIdx.x * blockDim.x + threadIdx.x;
    if (i < e) norm[i] = dinv[(int)src[i]] * dinv[(int)dst[i]];
}

// ---------------------------------------------------------------------------
// H[M,64] = X[M,128] @ W[128,64] using V_WMMA_F32_16X16X4_F32.
// One wave per 16x16 tile of H. K stepped by 4.
// A 16x4 layout : lanes 0-15 hold K=0,1 ; lanes 16-31 hold K=2,3 (VGPR0/1)
// B 4x16 layout : VGPR0 = rows K=0 (lanes 0-15) / K=2 (lanes 16-31), VGPR1 = K=1/K=3
// C/D layout    : VGPR v -> row (v + 8*half), col = lane&15
__global__ void k_gemm1_wmma(const float* __restrict__ X, const float* __restrict__ W,
                             float* __restrict__ H, int M) {
    int gwave  = (blockIdx.x * blockDim.x + threadIdx.x) >> 5;
    int lane   = threadIdx.x & 31;
    int tile_n = gwave & 3;    // 64 / 16 = 4 column tiles
    int tile_m = gwave >> 2;
    if (tile_m * 16 >= M) return;                 // wave-uniform, EXEC stays full

    int half = lane >> 4;      // 0 or 1
    int l15  = lane & 15;
    int arow = tile_m * 16 + l15;
    if (arow >= M) arow = M - 1;                  // clamp to keep EXEC all-ones
    int bcol = tile_n * 16 + l15;

    v8f c = {};
    #pragma unroll 4
    for (int k = 0; k < 128; k += 4) {
        int ka = k + half * 2;
        v2f a, b;
        a.x = X[arow * 128 + ka];
        a.y = X[arow * 128 + ka + 1];
        b.x = W[ka * 64 + bcol];
        b.y = W[(ka + 1) * 64 + bcol];
        c = __builtin_amdgcn_wmma_f32_16x16x4_f32(false, a, false, b, (short)0, c, false, false);
    }

    int orow = tile_m * 16 + half * 8;
    int ocol = tile_n * 16 + l15;
    if (tile_m * 16 + 16 <= M) {
        // full tile (always the case when M % 16 == 0): no per-row guards,
        // EXEC stays all-ones, plain 8x global_store_b32
        #pragma unroll
        for (int v = 0; v < 8; ++v)
            H[(orow + v) * 64 + ocol] = c[v];
    } else {
        // partial last tile (not taken for M = 100000)
        #pragma unroll
        for (int v = 0; v < 8; ++v) {
            int r = orow + v;
            if (r < M) H[r * 64 + ocol] = c[v];
        }
    }
}

// ---------------------------------------------------------------------------
// OUT[i,f] = H[i,f]*dinv[i]^2   (self-loop contribution; also zero-inits OUT)
__global__ void k_agg_init64(const float* __restrict__ H, const float* __restrict__ dinv,
                             float* __restrict__ OUT, int n) {
    int i = blockIdx.x * blockDim.x + threadIdx.x;   // over n*64
    if (i < n * 64) {
        int node = i >> 6;
        float d = dinv[node];
        OUT[i] = H[i] * d * d;
    }
}

// One wave per edge: OUT[dst, lane] += H[src, lane]*norm  (lane and lane+32)
__global__ void k_agg_edges64(const long long* __restrict__ src, const long long* __restrict__ dst,
                              const float* __restrict__ norm, const float* __restrict__ H,
                              float* __restrict__ OUT, int e) {
    int gwave = (blockIdx.x * (blockDim.x >> 5)) + (threadIdx.x >> 5);
    int lane  = threadIdx.x & 31;
    if (gwave >= e) return;
    int s = (int)src[gwave];
    int d = (int)dst[gwave];
    float nv = norm[gwave];
    atomicAdd(&OUT[d * 64 + lane],      H[s * 64 + lane]      * nv);
    atomicAdd(&OUT[d * 64 + lane + 32], H[s * 64 + lane + 32] * nv);
}

// A[i,f] = relu(B[i,f] + b1[f])
__global__ void k_bias_relu(const float* __restrict__ B, const float* __restrict__ b1,
                            float* __restrict__ A, int n) {
    int i = blockIdx.x * blockDim.x + threadIdx.x;
    if (i < n * 64) {
        float v = B[i] + b1[i & 63];
        A[i] = v > 0.0f ? v : 0.0f;
    }
}

// h2[i] = dot(H1[i, 0:64], W2[0:64])  — one wave per node, shuffle reduce
__global__ void k_gemm2(const float* __restrict__ H1, const float* __restrict__ W2,
                        float* __restrict__ H2, int n) {
    int gwave = (blockIdx.x * (blockDim.x >> 5)) + (threadIdx.x >> 5);
    int lane  = threadIdx.x & 31;
    if (gwave >= n) return;
    float s = H1[gwave * 64 + lane] * W2[lane] + H1[gwave * 64 + lane + 32] * W2[lane + 32];
    #pragma unroll
    for (int off = 16; off > 0; off >>= 1) s += __shfl_xor(s, off, 32);
    if (lane == 0) H2[gwave] = s;
}

// a2[i] = h2[i]*dinv[i]^2  (self-loop + init)
__global__ void k_agg_init1(const float* __restrict__ H2, const float* __restrict__ dinv,
                            float* __restrict__ OUT, int n) {
    int i = blockIdx.x * blockDim.x + threadIdx.x;
    if (i < n) {
        float d = dinv[i];
        OUT[i] = H2[i] * d * d;
    }
}

// thread per edge: OUT[dst] += h2[src]*norm
__global__ void k_agg_edges1(const long long* __restrict__ src, const long long* __restrict__ dst,
                             const float* __restrict__ norm, const float* __restrict__ H2,
                             float* __restrict__ OUT, int e) {
    int i = blockIdx.x * blockDim.x + threadIdx.x;
    if (i < e) atomicAdd(&OUT[(int)dst[i]], H2[(int)src[i]] * norm[i]);
}

// out[i] = sigmoid(a2[i] + b2)
__global__ void k_sigmoid(const float* __restrict__ A, const float* __restrict__ b2,
                          float* __restrict__ OUT, int n) {
    int i = blockIdx.x * blockDim.x + threadIdx.x;
    if (i < n) {
        float z = A[i] + b2[0];
        OUT[i] = 1.0f / (1.0f + __expf(-z));
    }
}

// ---------------------------------------------------------------------------
extern "C" void kernel_launch(void* const* d_in, const int* in_sizes, int n_in,
                              void* d_out, int out_size, void* d_ws, size_t ws_size,
                              hipStream_t stream) {
    const float*      x    = (const float*)d_in[0];       // [N,128]
    const long long*  ei   = (const long long*)d_in[1];   // [2,E] int64
    const float*      W1   = (const float*)d_in[2];       // [128,64]
    const float*      b1   = (const float*)d_in[3];       // [64]
    const float*      W2   = (const float*)d_in[4];       // [64,1]
    const float*      b2   = (const float*)d_in[5];       // [1]
    float*            out  = (float*)d_out;               // [N,1]

    const int N = in_sizes[0] / 128;
    const int E = in_sizes[1] / 2;
    const long long* src = ei;
    const long long* dst = ei + E;

    // workspace partition (floats)
    float* ws   = (float*)d_ws;
    float* dinv = ws;                      // N
    float* norm = dinv + N;                // E
    float* bufA = norm + E;                // N*64  (h, then relu(h_agg))
    float* bufB = bufA + (size_t)N * 64;   // N*64  (agg1 result)
    float* h2   = bufB + (size_t)N * 64;   // N
    float* a2   = h2 + N;                  // N

    const int B = 256;
    const int wavesPerBlock = B / WAVE;

    // degrees + normalization
    k_deg_init <<<(N + B - 1) / B, B, 0, stream>>>(dinv, N);
    k_deg_accum<<<(E + B - 1) / B, B, 0, stream>>>(dst, dinv, E);
    k_dinv     <<<(N + B - 1) / B, B, 0, stream>>>(dinv, N);
    k_norm     <<<(E + B - 1) / B, B, 0, stream>>>(src, dst, dinv, norm, E);

    // layer 1: GEMM (WMMA f32) -> aggregate -> bias+relu
    int tilesM = (N + 15) / 16;
    int waves1 = tilesM * 4;
    k_gemm1_wmma<<<(waves1 + wavesPerBlock - 1) / wavesPerBlock, B, 0, stream>>>(x, W1, bufA, N);

    k_agg_init64 <<<((size_t)N * 64 + B - 1) / B, B, 0, stream>>>(bufA, dinv, bufB, N);
    k_agg_edges64<<<(E + wavesPerBlock - 1) / wavesPerBlock, B, 0, stream>>>(src, dst, norm, bufA, bufB, E);
    k_bias_relu  <<<((size_t)N * 64 + B - 1) / B, B, 0, stream>>>(bufB, b1, bufA, N);

    // layer 2: dot product -> aggregate -> sigmoid
    k_gemm2     <<<(N + wavesPerBlock - 1) / wavesPerBlock, B, 0, stream>>>(bufA, W2, h2, N);
    k_agg_init1 <<<(N + B - 1) / B, B, 0, stream>>>(h2, dinv, a2, N);
    k_agg_edges1<<<(E + B - 1) / B, B, 0, stream>>>(src, dst, norm, h2, a2, E);
    k_sigmoid   <<<(N + B - 1) / B, B, 0, stream>>>(a2, b2, out, N);
}